// ContrastiveGAT_6433861009605
// MI455X (gfx1250) — compile-verified
//
#include <hip/hip_runtime.h>

// ---------------------------------------------------------------------------
// 2-layer GAT forward for MI455X (gfx1250, wave32).
//   layer1: x[N,256] @ W1[256,128] -> 4 heads x 32ch, segment-softmax, ELU
//   layer2: h[N,128] @ W2[128,128] -> 1 head x 128ch, segment-softmax, +b2
// GEMM: V_WMMA_F32_16X16X4_F32, one wave per 16x128 row strip (8 accumulator
// tiles, A fragment reused 8x). W k-panels staged in LDS in k-pair-interleaved
// layout: element (k,c) at (k/2)*288 + 2c + (k&1), so each B fragment is one
// aligned, bank-conflict-free ds_load_b64 straight into the WMMA operand.
// Edge phase: wave-per-edge float4 gather + global_atomic_add_f32 scatter;
// segment-max via order-preserving uint encoding + atomicMax (memset-0 init
// valid since every dst has a self loop). Whole working set (~135 MB) is
// L2-resident (192 MB L2, 23.3 TB/s HBM never touched after first pass).
// ---------------------------------------------------------------------------

typedef float v2f __attribute__((ext_vector_type(2)));
typedef float v8f __attribute__((ext_vector_type(8)));

#define KPANEL 64          // W k-rows staged per LDS panel (32 pair-rows)
#define PAIR_STRIDE 288    // floats per pair-row: 256 + 32 -> halves of the
                           // wave hit banks 0-31 / 32-63 (conflict-free b64)

// ---------------- fp32 WMMA GEMM: XP[N,128] = X[N,K] @ W[K,128] -------------
// Block: 256 threads = 8 waves; wave w computes rows [blk*128 + w*16, +16).
//   A tile 16x4 (f32): lanes 0-15 row M=lane  K={k,k+1};
//                      lanes 16-31 row M=lane-16 K={k+2,k+3}
//   B tile 4x16 (f32): VGPR0 rows k/k+2, VGPR1 rows k+1/k+3, col c0+(lane&15)
//   C/D 16x16 (8 VGPRs): VGPR v -> M=v (lanes 0-15), v+8 (lanes 16-31)
// Requires N % 16 == 0 (50000 = 16*3125) and K % KPANEL == 0 (256, 128).
__global__ void gat_gemm_wmma(const float* __restrict__ X,
                              const float* __restrict__ W,
                              float* __restrict__ XP,
                              int K, int N) {
  __shared__ float wlds[(KPANEL / 2) * PAIR_STRIDE];   // 36 KB

  const int tid  = threadIdx.x;        // 0..255
  const int lane = tid & 31;
  const int wid  = tid >> 5;           // 0..7
  const int half = lane >> 4;          // 0 or 1
  const int l15  = lane & 15;
  const int r0   = blockIdx.x * 128 + wid * 16;
  const bool active = (r0 < N);        // wave-uniform (N % 16 == 0)

  const float* __restrict__ xrow = X + (size_t)(r0 + l15) * K;

  v8f acc[8] = {};

  for (int kp = 0; kp < K; kp += KPANEL) {
    // ---- cooperative stage: W[kp..kp+64) x 128 -> LDS, k-pair interleaved.
    // Task t: pair-row p = t>>5, float4 col-group c4 = t&31. Reads one float4
    // from W rows kp+2p and kp+2p+1, writes them interleaved so (k,c),(k+1,c)
    // are adjacent -> B fragments become single ds_load_b64's.
    #pragma unroll
    for (int i = 0; i < 4; ++i) {                       // 1024 tasks / 256 thr
      const int t  = tid + i * 256;
      const int p  = t >> 5;           // 0..31
      const int c4 = t & 31;           // float4 group: cols 4c4..4c4+3
      const float4 a0 = ((const float4*)(W + (size_t)(kp + 2 * p) * 128))[c4];
      const float4 a1 = ((const float4*)(W + (size_t)(kp + 2 * p + 1) * 128))[c4];
      float4* dst = (float4*)(wlds + p * PAIR_STRIDE + 8 * c4);
      dst[0] = make_float4(a0.x, a1.x, a0.y, a1.y);
      dst[1] = make_float4(a0.z, a1.z, a0.w, a1.w);
    }
    __syncthreads();

    if (active) {
      // pull next A panel toward the WGP while we compute this one
      if (kp + KPANEL < K) __builtin_prefetch(xrow + kp + KPANEL, 0, 3);

      for (int k = 0; k < KPANEL; k += 4) {
        const int p = (k >> 1) + half;           // pair-row holding ka, ka+1
        v2f a;
        a.x = xrow[kp + k + 2 * half + 0];
        a.y = xrow[kp + k + 2 * half + 1];
        const float* __restrict__ brow = wlds + p * PAIR_STRIDE + 2 * l15;
        #pragma unroll
        for (int ct = 0; ct < 8; ++ct) {         // 8 column tiles, A reused
          const v2f b = *(const v2f*)(brow + ct * 32);   // (ka,c),(ka+1,c)
          acc[ct] = __builtin_amdgcn_wmma_f32_16x16x4_f32(
              /*neg_a=*/false, a, /*neg_b=*/false, b,
              /*c_mod=*/(short)0, acc[ct], /*reuse_a=*/false, /*reuse_b=*/false);
        }
      }
    }
    __syncthreads();
  }

  if (active) {
    #pragma unroll
    for (int ct = 0; ct < 8; ++ct) {
      #pragma unroll
      for (int v = 0; v < 8; ++v) {
        XP[(size_t)(r0 + v + 8 * half) * 128 + ct * 16 + l15] = acc[ct][v];
      }
    }
  }
}

// ---------------- per-node attention logits --------------------------------
// al_s[n,h] = sum_c xp[n,h,c]*a_src[h,c] ; al_d likewise. C = 128/H.
template <int H>
__global__ void gat_node_logits(const float* __restrict__ XP,
                                const float* __restrict__ a_src,
                                const float* __restrict__ a_dst,
                                float* __restrict__ al_s,
                                float* __restrict__ al_d,
                                int n_nodes) {
  const int t = blockIdx.x * blockDim.x + threadIdx.x;
  if (t >= n_nodes * H) return;
  constexpr int C = 128 / H;
  const int n = t / H, h = t % H;
  const float* __restrict__ xv = XP + (size_t)n * 128 + h * C;
  const float* __restrict__ as = a_src + h * C;
  const float* __restrict__ ad = a_dst + h * C;
  float ss = 0.f, sd = 0.f;
  #pragma unroll 8
  for (int c = 0; c < C; ++c) {
    const float x = xv[c];
    ss += x * as[c];
    sd += x * ad[c];
  }
  al_s[t] = ss;
  al_d[t] = sd;
}

// Order-preserving float <-> uint encoding (total order, matches float <).
__device__ __forceinline__ unsigned f32_ord(float f) {
  unsigned u = __float_as_uint(f);
  return u ^ ((u >> 31) ? 0xFFFFFFFFu : 0x80000000u);
}
__device__ __forceinline__ float ord_f32(unsigned x) {
  const unsigned u = x ^ ((x >> 31) ? 0x80000000u : 0xFFFFFFFFu);
  return __uint_as_float(u);
}
__device__ __forceinline__ float leaky(float v) { return v > 0.f ? v : 0.2f * v; }

// ---------------- edge pass 1: segment max over dst ------------------------
template <int H>
__global__ void gat_edge_max(const int* __restrict__ src,
                             const int* __restrict__ dst,
                             const float* __restrict__ al_s,
                             const float* __restrict__ al_d,
                             unsigned* __restrict__ m_enc,
                             int E, int n_nodes) {
  const int e = blockIdx.x * blockDim.x + threadIdx.x;
  if (e >= E + n_nodes) return;
  const int s = (e < E) ? src[e] : (e - E);   // self loops appended
  const int d = (e < E) ? dst[e] : (e - E);
  #pragma unroll
  for (int h = 0; h < H; ++h) {
    const float v = leaky(al_s[s * H + h] + al_d[d * H + h]);
    atomicMax(&m_enc[d * H + h], f32_ord(v));
  }
}

// ---------------- edge pass 2: segment exp-sum -----------------------------
template <int H>
__global__ void gat_edge_expsum(const int* __restrict__ src,
                                const int* __restrict__ dst,
                                const float* __restrict__ al_s,
                                const float* __restrict__ al_d,
                                const unsigned* __restrict__ m_enc,
                                float* __restrict__ z,
                                int E, int n_nodes) {
  const int e = blockIdx.x * blockDim.x + threadIdx.x;
  if (e >= E + n_nodes) return;
  const int s = (e < E) ? src[e] : (e - E);
  const int d = (e < E) ? dst[e] : (e - E);
  #pragma unroll
  for (int h = 0; h < H; ++h) {
    const float v = leaky(al_s[s * H + h] + al_d[d * H + h]);
    atomicAdd(&z[d * H + h], __expf(v - ord_f32(m_enc[d * H + h])));
  }
}

// ---------------- edge pass 3: alpha * xp[src] scattered to acc[dst] -------
// One wave per edge; each lane handles 4 contiguous channels (coalesced
// 512B gather per edge, L2-resident), 4 global_atomic_add_f32 per lane.
template <int H>
__global__ void gat_edge_aggr(const int* __restrict__ src,
                              const int* __restrict__ dst,
                              const float* __restrict__ al_s,
                              const float* __restrict__ al_d,
                              const unsigned* __restrict__ m_enc,
                              const float* __restrict__ z,
                              const float* __restrict__ XP,
                              float* __restrict__ acc,
                              int E, int n_nodes) {
  const int wid  = (blockIdx.x * blockDim.x + threadIdx.x) >> 5;  // edge id
  const int lane = threadIdx.x & 31;
  if (wid >= E + n_nodes) return;
  const int s = (wid < E) ? src[wid] : (wid - E);
  const int d = (wid < E) ? dst[wid] : (wid - E);
  constexpr int C = 128 / H;
  const int c4 = lane * 4;
  const int h  = c4 / C;                 // channels 4l..4l+3 share one head
  const float v  = leaky(al_s[s * H + h] + al_d[d * H + h]);
  const float ex = __expf(v - ord_f32(m_enc[d * H + h]));
  const float alpha = ex / (z[d * H + h] + 1e-16f);
  const float4 xv = ((const float4*)XP)[(size_t)s * 32 + lane];
  float* o = acc + (size_t)d * 128 + c4;
  atomicAdd(o + 0, alpha * xv.x);
  atomicAdd(o + 1, alpha * xv.y);
  atomicAdd(o + 2, alpha * xv.z);
  atomicAdd(o + 3, alpha * xv.w);
}

// ---------------- finalize: add bias, optional ELU, in place ---------------
__global__ void gat_finalize(float* __restrict__ acc,
                             const float* __restrict__ bias,
                             int n_nodes, int do_elu) {
  const int t = blockIdx.x * blockDim.x + threadIdx.x;
  if (t >= n_nodes * 128) return;
  float v = acc[t] + bias[t & 127];
  if (do_elu) v = (v > 0.f) ? v : (__expf(v) - 1.f);
  acc[t] = v;
}

// ---------------------------------------------------------------------------
extern "C" void kernel_launch(void* const* d_in, const int* in_sizes, int n_in,
                              void* d_out, int out_size, void* d_ws, size_t ws_size,
                              hipStream_t stream) {
  const float* x   = (const float*)d_in[0];
  const int*   src = (const int*)  d_in[1];
  const int*   dst = (const int*)  d_in[2];
  const float* W1  = (const float*)d_in[3];
  const float* as1 = (const float*)d_in[4];
  const float* ad1 = (const float*)d_in[5];
  const float* b1  = (const float*)d_in[6];
  const float* W2  = (const float*)d_in[7];
  const float* as2 = (const float*)d_in[8];
  const float* ad2 = (const float*)d_in[9];
  const float* b2  = (const float*)d_in[10];

  const int IN_CH = 256;                 // layer-1 input width
  const int N    = in_sizes[0] / IN_CH;  // 50000 (multiple of 16)
  const int E    = in_sizes[1];          // 800000
  const int Etot = E + N;                // + self loops

  // workspace layout (floats): ~80 MB
  float*    xp1   = (float*)d_ws;                      // N*128
  float*    h     = xp1 + (size_t)N * 128;             // acc1, then hidden
  float*    xp2   = h   + (size_t)N * 128;             // N*128
  float*    al_s  = xp2 + (size_t)N * 128;             // N*4 (max heads)
  float*    al_d  = al_s + (size_t)N * 4;              // N*4
  unsigned* m_enc = (unsigned*)(al_d + (size_t)N * 4); // N*4
  float*    z     = (float*)(m_enc + (size_t)N * 4);   // N*4
  float*    out   = (float*)d_out;

  const int GB = (N + 127) / 128;        // GEMM blocks (128 rows each)
  const int EB = (Etot + 255) / 256;     // edge-grid (thread per edge)
  const int WB = (Etot + 7) / 8;         // edge-grid (wave per edge)
  const int NB = (N * 128 + 255) / 256;

  // -------- layer 1: 4 heads x 32 channels --------
  gat_gemm_wmma<<<GB, 256, 0, stream>>>(x, W1, xp1, IN_CH, N);
  gat_node_logits<4><<<(N * 4 + 255) / 256, 256, 0, stream>>>(xp1, as1, ad1, al_s, al_d, N);
  hipMemsetAsync(h,     0, (size_t)N * 128 * sizeof(float), stream);
  hipMemsetAsync(m_enc, 0, (size_t)N * 4 * sizeof(unsigned), stream);
  hipMemsetAsync(z,     0, (size_t)N * 4 * sizeof(float), stream);
  gat_edge_max<4>   <<<EB, 256, 0, stream>>>(src, dst, al_s, al_d, m_enc, E, N);
  gat_edge_expsum<4><<<EB, 256, 0, stream>>>(src, dst, al_s, al_d, m_enc, z, E, N);
  gat_edge_aggr<4>  <<<WB, 256, 0, stream>>>(src, dst, al_s, al_d, m_enc, z, xp1, h, E, N);
  gat_finalize<<<NB, 256, 0, stream>>>(h, b1, N, /*elu=*/1);

  // -------- layer 2: 1 head x 128 channels --------
  gat_gemm_wmma<<<GB, 256, 0, stream>>>(h, W2, xp2, 128, N);
  gat_node_logits<1><<<(N + 255) / 256, 256, 0, stream>>>(xp2, as2, ad2, al_s, al_d, N);
  hipMemsetAsync(out,   0, (size_t)N * 128 * sizeof(float), stream);
  hipMemsetAsync(m_enc, 0, (size_t)N * sizeof(unsigned), stream);
  hipMemsetAsync(z,     0, (size_t)N * sizeof(float), stream);
  gat_edge_max<1>   <<<EB, 256, 0, stream>>>(src, dst, al_s, al_d, m_enc, E, N);
  gat_edge_expsum<1><<<EB, 256, 0, stream>>>(src, dst, al_s, al_d, m_enc, z, E, N);
  gat_edge_aggr<1>  <<<WB, 256, 0, stream>>>(src, dst, al_s, al_d, m_enc, z, xp2, out, E, N);
  gat_finalize<<<NB, 256, 0, stream>>>(out, b2, N, /*elu=*/0);
}